// MVDRBeamformer_11982958756551
// MI455X (gfx1250) — compile-verified
//
#include <hip/hip_runtime.h>
#include <math.h>

typedef float v2f __attribute__((ext_vector_type(2)));
typedef float v8f __attribute__((ext_vector_type(8)));

#define NUM_B 8
#define NUM_C 4
#define NUM_T 600
#define NUM_F 257
#define ALPHA 0.14785621f
#define OMA   (1.0f - ALPHA)
#define EPSV  1e-12f

// fast (~1 ulp) reciprocal / sqrt: the reference regularizes every division
// with 1e-8..1e-12 epsilons, so v_rcp_f32 accuracy is more than sufficient,
// and it avoids the ~8-instruction IEEE div_scale/div_fmas expansion on the
// latency-critical per-time-step solve.
__device__ __forceinline__ float frcp(float x)  { return __builtin_amdgcn_rcpf(x); }
__device__ __forceinline__ float fsqrt(float x) { return __builtin_amdgcn_sqrtf(x); }

struct cplx { float re, im; };
__device__ __forceinline__ cplx cmul(cplx a, cplx b){ return {a.re*b.re - a.im*b.im, a.re*b.im + a.im*b.re}; }
__device__ __forceinline__ cplx cconjf(cplx a){ return {a.re, -a.im}; }
__device__ __forceinline__ cplx cadd(cplx a, cplx b){ return {a.re+b.re, a.im+b.im}; }
__device__ __forceinline__ cplx csub(cplx a, cplx b){ return {a.re-b.re, a.im-b.im}; }
__device__ __forceinline__ cplx cinvf(cplx a){
    const float id = frcp(a.re*a.re + a.im*a.im + 1e-30f);
    return {a.re*id, -a.im*id};
}

// Build the 32 packed real columns for frame t:
//  cols 0..15 : n * Hermitian(M)  (4 real diag, 6 complex off-diag)
//  cols 16..22: s * M[:,0]        (Rxx first column; M00 real)
//  cols 23..31: zero padding
__device__ __forceinline__ void frame_cols(const float* __restrict__ xr,
                                           const float* __restrict__ xi,
                                           const float* __restrict__ mask,
                                           int b, int f, int t,
                                           float* v, float* mre, float* mim)
{
    const size_t midx = (((size_t)b*NUM_T + t)*NUM_F + f)*2;
    const float m_r = mask[midx + 0];
    const float m_i = mask[midx + 1];
    const float mag = fsqrt(m_r*m_r + m_i*m_i);
    const float pinv = frcp(mag + 1e-8f);
    const float pr = m_r * pinv, pi = m_i * pinv;
    float g = (mag < 0.3f) ? 0.0f : mag;
    g = (g > 0.7f) ? 1.0f : g;
    const float cr = g*pr, ci = g*pi;
    const float s = cr*cr + ci*ci;                         // |cRM|^2
    const float n = (1.0f-cr)*(1.0f-cr) + ci*ci;           // |1-cRM|^2

#pragma unroll
    for (int c = 0; c < NUM_C; ++c) {
        const size_t idx = (((size_t)b*NUM_C + c)*NUM_T + t)*NUM_F + f;
        mre[c] = xr[idx];
        mim[c] = xi[idx];
    }
    // diagonal of M
    v[0] = n*(mre[0]*mre[0]+mim[0]*mim[0]);
    v[1] = n*(mre[1]*mre[1]+mim[1]*mim[1]);
    v[2] = n*(mre[2]*mre[2]+mim[2]*mim[2]);
    v[3] = n*(mre[3]*mre[3]+mim[3]*mim[3]);
    // off-diagonals M[c][d] = m_c * conj(m_d)
    const int pc[6] = {1,2,3,2,3,3};
    const int pd[6] = {0,0,0,1,1,2};
#pragma unroll
    for (int k = 0; k < 6; ++k) {
        const int c = pc[k], d = pd[k];
        const float ore = mre[c]*mre[d] + mim[c]*mim[d];
        const float oim = mim[c]*mre[d] - mre[c]*mim[d];
        v[4 + 2*k]     = n*ore;
        v[4 + 2*k + 1] = n*oim;
    }
    // Rxx first column (s * M[:,0])
    v[16] = s*(mre[0]*mre[0]+mim[0]*mim[0]);
#pragma unroll
    for (int c = 1; c < 4; ++c) {
        const float ore = mre[c]*mre[0] + mim[c]*mim[0];
        const float oim = mim[c]*mre[0] - mre[c]*mim[0];
        v[16 + 2*c - 1] = s*ore;
        v[16 + 2*c]     = s*oim;
    }
#pragma unroll
    for (int k = 23; k < 32; ++k) v[k] = 0.0f;
}

__global__ __launch_bounds__(32)
void mvdr_wmma_kernel(const float* __restrict__ xr, const float* __restrict__ xi,
                      const float* __restrict__ mask, float* __restrict__ out)
{
    __shared__ float sR[16][32];     // frame rows (time x packed cols)
    __shared__ float sS[16][32];     // smoothed rows
    __shared__ float sCarry[32];     // EMA carry (last smoothed row)

    const int bf   = blockIdx.x;
    const int b    = bf / NUM_F;
    const int f    = bf % NUM_F;
    const int lane = (int)threadIdx.x;
    const int col  = lane & 15;      // N index / A-matrix row
    const int half = lane >> 4;      // 0: lanes 0-15, 1: lanes 16-31

    // ---- constant WMMA A operand: W[i][j] = alpha*(1-a)^(i-j), lower-tri 16x16
    // A 16x4 f32 layout: lane m=col holds K = (kbase + 2*half) in .x, +1 in .y
    v2f wa[4];
#pragma unroll
    for (int c4 = 0; c4 < 4; ++c4) {
        const int j0 = c4*4 + 2*half;
        wa[c4].x = (j0   <= col) ? (ALPHA * powf(OMA, (float)(col - j0)))     : 0.0f;
        wa[c4].y = (j0+1 <= col) ? (ALPHA * powf(OMA, (float)(col - j0 - 1))) : 0.0f;
    }
    // carry coefficients (1-a)^(M+1) for this lane's 8 accumulator rows
    float coef[8];
#pragma unroll
    for (int r = 0; r < 8; ++r) coef[r] = powf(OMA, (float)(r + 8*half + 1));

    float mre[4], mim[4];            // cached mic samples for this lane's time step

    // ---- seed carry with R[0] so S[0] = (1-a)R[0] + aR[0] = R[0]
    if (lane == 0) {
        float tr[4], ti[4];
        frame_cols(xr, xi, mask, b, f, 0, &sCarry[0], tr, ti);
    }
    __syncthreads();

    const int NTILE = (NUM_T + 15) / 16;   // 38 tiles
    for (int tt = 0; tt < NTILE; ++tt) {
        const int t0 = tt * 16;

        // phase 1: build 16 frame rows into LDS (one lane per time step),
        // then prefetch next tile's strided inputs (hidden under WMMA/solve).
        if (lane < 16) {
            const int t = t0 + lane;
            if (t < NUM_T) {
                frame_cols(xr, xi, mask, b, f, t, &sR[lane][0], mre, mim);
            } else {
#pragma unroll
                for (int k = 0; k < 32; ++k) sR[lane][k] = 0.0f;
            }
            const int tn = t + 16;
            if (tn < NUM_T) {
                __builtin_prefetch(&mask[(((size_t)b*NUM_T + tn)*NUM_F + f)*2], 0, 0);
#pragma unroll
                for (int c = 0; c < NUM_C; ++c) {
                    const size_t idx = (((size_t)b*NUM_C + c)*NUM_T + tn)*NUM_F + f;
                    __builtin_prefetch(&xr[idx], 0, 0);
                    __builtin_prefetch(&xi[idx], 0, 0);
                }
            }
        }
        __syncthreads();

        // phase 2: S = W * R + diag(coef) * carry   via 8x v_wmma_f32_16x16x4_f32
        v8f acc0, acc1;
#pragma unroll
        for (int r = 0; r < 8; ++r) {
            acc0[r] = coef[r] * sCarry[col];
            acc1[r] = coef[r] * sCarry[16 + col];
        }
#pragma unroll
        for (int c4 = 0; c4 < 4; ++c4) {
            const int j0 = c4*4 + 2*half;   // B 4x16 layout: .x row K=j0, .y row K=j0+1
            v2f b0, b1;
            b0.x = sR[j0][col];        b0.y = sR[j0+1][col];
            b1.x = sR[j0][16 + col];   b1.y = sR[j0+1][16 + col];
            acc0 = __builtin_amdgcn_wmma_f32_16x16x4_f32(false, wa[c4], false, b0,
                                                         (short)0, acc0, false, false);
            acc1 = __builtin_amdgcn_wmma_f32_16x16x4_f32(false, wa[c4], false, b1,
                                                         (short)0, acc1, false, false);
        }
        // phase 3: spill smoothed tile to LDS (D layout: M = r + 8*half, N = col)
#pragma unroll
        for (int r = 0; r < 8; ++r) {
            const int m = r + 8*half;
            sS[m][col]      = acc0[r];
            sS[m][16 + col] = acc1[r];
        }
        __syncthreads();

        // phase 4: carry = last smoothed row
        sCarry[lane] = sS[15][lane];

        // phase 5: MVDR solve + beamform, one lane per time step
        if (lane < 16) {
            const int t = t0 + lane;
            if (t < NUM_T) {
                const float* r = &sS[lane][0];
                // reconstruct Rnn (Hermitian) + 1e-9*I
                cplx A[4][4];
                A[0][0] = {r[0] + 1e-9f, 0.0f};
                A[1][1] = {r[1] + 1e-9f, 0.0f};
                A[2][2] = {r[2] + 1e-9f, 0.0f};
                A[3][3] = {r[3] + 1e-9f, 0.0f};
                A[1][0] = {r[4],  r[5]};  A[0][1] = cconjf(A[1][0]);
                A[2][0] = {r[6],  r[7]};  A[0][2] = cconjf(A[2][0]);
                A[3][0] = {r[8],  r[9]};  A[0][3] = cconjf(A[3][0]);
                A[2][1] = {r[10], r[11]}; A[1][2] = cconjf(A[2][1]);
                A[3][1] = {r[12], r[13]}; A[1][3] = cconjf(A[3][1]);
                A[3][2] = {r[14], r[15]}; A[2][3] = cconjf(A[3][2]);

                // rtf: Rxx[:,0]/(Rxx00+eps), phase-only
                const float invx = frcp(r[16] + EPSV);
                cplx rtf[4];
                rtf[0] = {r[16]*invx, 0.0f};
                rtf[1] = {r[17]*invx, r[18]*invx};
                rtf[2] = {r[19]*invx, r[20]*invx};
                rtf[3] = {r[21]*invx, r[22]*invx};
#pragma unroll
                for (int c = 0; c < 4; ++c) {
                    const float m2 = fsqrt(rtf[c].re*rtf[c].re + rtf[c].im*rtf[c].im);
                    const float s2 = frcp(m2 + 1e-8f);
                    rtf[c].re *= s2; rtf[c].im *= s2;
                }
                // solve A x = rtf (Gauss, no pivot: Hermitian + regularized)
                cplx bv[4] = {rtf[0], rtf[1], rtf[2], rtf[3]};
#pragma unroll
                for (int k = 0; k < 4; ++k) {
                    const cplx piv = cinvf(A[k][k]);
#pragma unroll
                    for (int j = 0; j < 4; ++j) if (j >= k) A[k][j] = cmul(A[k][j], piv);
                    bv[k] = cmul(bv[k], piv);
#pragma unroll
                    for (int i = 0; i < 4; ++i) if (i > k) {
                        const cplx fac = A[i][k];
#pragma unroll
                        for (int j = 0; j < 4; ++j) if (j >= k) A[i][j] = csub(A[i][j], cmul(fac, A[k][j]));
                        bv[i] = csub(bv[i], cmul(fac, bv[k]));
                    }
                }
                cplx x[4];
#pragma unroll
                for (int k = 3; k >= 0; --k) {
                    cplx sm = bv[k];
#pragma unroll
                    for (int j = 0; j < 4; ++j) if (j > k) sm = csub(sm, cmul(A[k][j], x[j]));
                    x[k] = sm;
                }
                // den = rtf^H x ; w = x/(den+eps) ; clean = w^H m
                cplx den = {EPSV, 0.0f};
#pragma unroll
                for (int c = 0; c < 4; ++c) den = cadd(den, cmul(cconjf(rtf[c]), x[c]));
                const cplx dinv = cinvf(den);
                cplx accv = {0.0f, 0.0f};
#pragma unroll
                for (int c = 0; c < 4; ++c) {
                    const cplx w = cmul(x[c], dinv);
                    accv = cadd(accv, cmul(cconjf(w), cplx{mre[c], mim[c]}));
                }
                const size_t oidx = (((size_t)b*NUM_T + t)*NUM_F + f)*2;
                out[oidx]     = accv.re;
                out[oidx + 1] = accv.im;
            }
        }
        __syncthreads();
    }
}

extern "C" void kernel_launch(void* const* d_in, const int* in_sizes, int n_in,
                              void* d_out, int out_size, void* d_ws, size_t ws_size,
                              hipStream_t stream) {
    const float* xr   = (const float*)d_in[0];
    const float* xi   = (const float*)d_in[1];
    const float* mask = (const float*)d_in[2];
    float* out        = (float*)d_out;
    (void)in_sizes; (void)n_in; (void)out_size; (void)d_ws; (void)ws_size;

    dim3 grid(NUM_B * NUM_F);   // 2056 independent (b,f) EMA chains, one wave32 each
    dim3 block(32);
    mvdr_wmma_kernel<<<grid, block, 0, stream>>>(xr, xi, mask, out);
}